// Sampler_70308614636114
// MI455X (gfx1250) — compile-verified
//
#include <hip/hip_runtime.h>
#include <stdint.h>

#define NB 64            // batch (rows sampled)
#define DD 2048          // hidden dim
#define VV 50257         // vocab
#define KC 128           // K-chunk per TDM tile (floats)
#define LROW 132         // padded LDS row stride in floats (128 + 4 pad -> bank-conflict free)
#define TILEF (64 * LROW)  // floats per (64-row x KC) LDS tile = 8448

typedef __attribute__((ext_vector_type(2))) float v2f;
typedef __attribute__((ext_vector_type(8))) float v8f;
typedef __attribute__((ext_vector_type(4))) unsigned int v4u;
typedef __attribute__((ext_vector_type(8))) int v8i;
typedef __attribute__((ext_vector_type(4))) int v4i;

// ---------------------------------------------------------------------------
// 1) Gather last-token hidden states and fold 1/temperature into them.
// ---------------------------------------------------------------------------
__global__ void gather_scale_kernel(const float* __restrict__ hs,
                                    const long long* __restrict__ idx,
                                    const float* __restrict__ temps,
                                    float* __restrict__ h) {
  const int b = blockIdx.x;
  const long long r = idx[b];
  const float t = temps[b];
  const float invt = (t < 1e-5f) ? 1.0f : (1.0f / t);
  const float* __restrict__ src = hs + r * (long long)DD;
  float* __restrict__ dst = h + (long long)b * DD;
  for (int d = threadIdx.x; d < DD; d += blockDim.x)
    dst[d] = src[d] * invt;
}

// ---------------------------------------------------------------------------
// TDM helper: DMA a (tile_d1 rows x tile_d0 elems) f32 tile into LDS with
// 4-dword padding every 128 dwords (LDS row stride = 132 dwords).
// D# layout per CDNA5 ISA 8.3/8.4; groups 2/3 (and the extra group) zero.
// ---------------------------------------------------------------------------
__device__ __forceinline__ void tdm_load_2d_f32(unsigned lds_byte_off,
                                                const float* gsrc,
                                                unsigned tensor_d0, unsigned tensor_d1,
                                                unsigned tile_d0, unsigned tile_d1,
                                                unsigned stride0) {
  const unsigned long long ga = (unsigned long long)(uintptr_t)gsrc;
  v4u g0;
  g0[0] = 1u;                                        // count=1, user mode
  g0[1] = lds_byte_off;                              // lds_addr
  g0[2] = (unsigned)(ga & 0xFFFFFFFFull);            // global_addr[31:0]
  g0[3] = (unsigned)((ga >> 32) & 0x01FFFFFFull)     // global_addr[56:32]
        | (2u << 30);                                // type = 2 (image)
  v8i g1;
  g1[0] = (int)((2u << 16)                           // data_size = 4B
              | (1u << 20)                           // pad_enable
              | (6u << 22)                           // pad_interval: 128 dwords
              | (3u << 25));                         // pad_amount: 4 dwords
  g1[1] = (int)((tensor_d0 & 0xFFFFu) << 16);        // ab_addr=0 | tensor_dim0[15:0]
  g1[2] = (int)(((tensor_d0 >> 16) & 0xFFFFu) | ((tensor_d1 & 0xFFFFu) << 16));
  g1[3] = (int)(((tensor_d1 >> 16) & 0xFFFFu) | ((tile_d0 & 0xFFFFu) << 16));
  g1[4] = (int)(tile_d1 & 0xFFFFu);                  // tile_dim1 | tile_dim2=0
  g1[5] = (int)stride0;                              // tensor_dim0_stride[31:0]
  g1[6] = 0;                                         // stride0 hi | stride1 lo
  g1[7] = 0;
  const v4i z4 = {0, 0, 0, 0};
  const v8i z8 = {0, 0, 0, 0, 0, 0, 0, 0};
  __builtin_amdgcn_tensor_load_to_lds(g0, g1, z4, z4, z8, 0);
}

__device__ __forceinline__ unsigned lds_off_of(const void* p) {
  return (unsigned)(unsigned long long)(uintptr_t)p;   // low 32 bits = LDS offset
}

// ---------------------------------------------------------------------------
// 2) logits[64 x V] = h[64 x 2048] @ E^T via V_WMMA_F32_16X16X4_F32.
//    Each block (4 waves) owns 64 vocab columns. TDM double-buffers
//    64x128 f32 tiles of E (wave0) and of h (wave1) into padded LDS;
//    waves compute from LDS with conflict-free ds_load_b64 fragments.
//    Pipeline: issue chunk c+1, s_wait_tensorcnt(1) for chunk c, barrier,
//    compute, barrier (fences buffer reuse against the next DMA).
// ---------------------------------------------------------------------------
__global__ void logits_wmma_kernel(const float* __restrict__ h,
                                   const float* __restrict__ E,
                                   float* __restrict__ logits) {
  extern __shared__ float smem[];       // layout: [A0 | A1 | B0 | B1], TILEF floats each

  int rb = (int)blockIdx.x * 64;
  if (rb > VV - 64) rb = VV - 64;                 // overlap-clamp: rewrites same values
  const int wv    = (int)(threadIdx.x >> 5);      // wave 0..3
  const int lane  = (int)(threadIdx.x & 31);
  const int mrow  = lane & 15;
  const int khalf = (lane >> 4) << 1;             // 0 or 2
  const int ncol  = rb + wv * 16 + mrow;          // always < VV after clamp

  const int NKC = DD / KC;                        // 16 chunks

  // prologue: chunk 0 in flight
  if (wv == 0) tdm_load_2d_f32(lds_off_of(smem + 2 * TILEF), E + (long long)rb * DD,
                               DD, VV, KC, 64, DD);
  if (wv == 1) tdm_load_2d_f32(lds_off_of(smem), h,
                               DD, 64, KC, 64, DD);

  v8f acc0 = {}; v8f acc1 = {}; v8f acc2 = {}; v8f acc3 = {};

  for (int kc = 0; kc < NKC; ++kc) {
    const int cur = kc & 1;
    const int nxt = cur ^ 1;
    if (wv == 0) {
      if (kc + 1 < NKC) {
        tdm_load_2d_f32(lds_off_of(smem + (2 + nxt) * TILEF),
                        E + (long long)rb * DD + (kc + 1) * KC, DD, VV, KC, 64, DD);
        __builtin_amdgcn_s_wait_tensorcnt(1);     // oldest (chunk kc) done
      } else {
        __builtin_amdgcn_s_wait_tensorcnt(0);     // final chunk done
      }
    }
    if (wv == 1) {
      if (kc + 1 < NKC) {
        tdm_load_2d_f32(lds_off_of(smem + nxt * TILEF),
                        h + (kc + 1) * KC, DD, 64, KC, 64, DD);
        __builtin_amdgcn_s_wait_tensorcnt(1);
      } else {
        __builtin_amdgcn_s_wait_tensorcnt(0);
      }
    }
    __syncthreads();                              // tile visible to all waves

    const float* __restrict__ Ap = smem + cur * TILEF + mrow * LROW + khalf;
    const float* __restrict__ Bp = smem + (2 + cur) * TILEF + (wv * 16 + mrow) * LROW + khalf;

#pragma unroll 4
    for (int ks = 0; ks < KC; ks += 4) {
      const v2f bf = *(const v2f*)(Bp + ks);
      const v2f a0 = *(const v2f*)(Ap + ks);
      const v2f a1 = *(const v2f*)(Ap + 16 * LROW + ks);
      const v2f a2 = *(const v2f*)(Ap + 32 * LROW + ks);
      const v2f a3 = *(const v2f*)(Ap + 48 * LROW + ks);
      acc0 = __builtin_amdgcn_wmma_f32_16x16x4_f32(false, a0, false, bf, (short)0, acc0, false, false);
      acc1 = __builtin_amdgcn_wmma_f32_16x16x4_f32(false, a1, false, bf, (short)0, acc1, false, false);
      acc2 = __builtin_amdgcn_wmma_f32_16x16x4_f32(false, a2, false, bf, (short)0, acc2, false, false);
      acc3 = __builtin_amdgcn_wmma_f32_16x16x4_f32(false, a3, false, bf, (short)0, acc3, false, false);
    }
    __syncthreads();                              // all reads done before next DMA lands
  }

  // C layout: VGPR r holds M=r (lanes 0-15) / M=r+8 (lanes 16-31), N = lane&15
  const int rbase = (lane >> 4) * 8;
  union { v8f v; float f[8]; } u0, u1, u2, u3;
  u0.v = acc0; u1.v = acc1; u2.v = acc2; u3.v = acc3;
#pragma unroll
  for (int r = 0; r < 8; ++r) {
    logits[(long long)(rbase + r) * VV + ncol]      = u0.f[r];
    logits[(long long)(16 + rbase + r) * VV + ncol] = u1.f[r];
    logits[(long long)(32 + rbase + r) * VV + ncol] = u2.f[r];
    logits[(long long)(48 + rbase + r) * VV + ncol] = u3.f[r];
  }
}

// ---------------------------------------------------------------------------
// 3) Sort-free top-p/top-k + softmax (dual 4-level radix descent; both masks
//    are prefix masks of the descending sort -> keep iff key >= max(tauK,tauP)).
// ---------------------------------------------------------------------------
__device__ __forceinline__ unsigned mkey(float f) {
  unsigned u = __float_as_uint(f);
  return (u & 0x80000000u) ? ~u : (u | 0x80000000u);
}

__global__ void sample_kernel(const float* __restrict__ logits,
                              const float* __restrict__ top_ps,
                              const long long* __restrict__ top_ks,
                              float* __restrict__ probs,
                              float* __restrict__ logprobs,
                              float* __restrict__ next_tok) {
  const int b = blockIdx.x;
  const int tid = (int)threadIdx.x;
  const int nt = (int)blockDim.x;        // 256
  const float* __restrict__ row = logits + (long long)b * VV;

  __shared__ float    redf[256];
  __shared__ int      redi[256];
  __shared__ unsigned s_cnt[256];
  __shared__ float    s_esum[256];
  __shared__ unsigned sh_kpfx, sh_ppfx, sh_krank;
  __shared__ float    sh_prem;

  const float NEGINF = -__builtin_inff();

  // ---- row max + first-occurrence argmax ----
  float m = NEGINF; int mi = 0;
  for (int v = tid; v < VV; v += nt) {
    const float x = row[v];
    if (x > m) { m = x; mi = v; }
  }
  redf[tid] = m; redi[tid] = mi;
  __syncthreads();
  for (int s = nt >> 1; s > 0; s >>= 1) {
    if (tid < s) {
      const float om = redf[tid + s]; const int oi = redi[tid + s];
      if (om > redf[tid] || (om == redf[tid] && oi < redi[tid])) { redf[tid] = om; redi[tid] = oi; }
    }
    __syncthreads();
  }
  const float rowmax = redf[0];
  const int   amax   = redi[0];
  __syncthreads();

  // ---- full partition function Z ----
  float zs = 0.f;
  for (int v = tid; v < VV; v += nt) zs += expf(row[v] - rowmax);
  redf[tid] = zs;
  __syncthreads();
  for (int s = nt >> 1; s > 0; s >>= 1) { if (tid < s) redf[tid] += redf[tid + s]; __syncthreads(); }
  const float Z = redf[0];
  __syncthreads();

  long long kk = top_ks[b];
  if (kk < 1)  kk = 1;
  if (kk > VV) kk = VV;
  if (tid == 0) {
    sh_kpfx = 0u; sh_ppfx = 0u;
    sh_krank = (unsigned)kk;
    sh_prem  = top_ps[b] * Z;
  }
  __syncthreads();

  // ---- dual radix descent, 8 bits/level ----
  for (int level = 0; level < 4; ++level) {
    const int shift = 24 - 8 * level;
    const unsigned kpfx = sh_kpfx, ppfx = sh_ppfx;
    s_cnt[tid] = 0u; s_esum[tid] = 0.f;
    __syncthreads();
    for (int v = tid; v < VV; v += nt) {
      const float x = row[v];
      const unsigned key = mkey(x);
      const unsigned bkt = (key >> shift) & 0xFFu;
      const bool inK = (level == 0) || ((key >> (shift + 8)) == kpfx);
      const bool inP = (level == 0) || ((key >> (shift + 8)) == ppfx);
      if (inK) atomicAdd(&s_cnt[bkt], 1u);
      if (inP) atomicAdd(&s_esum[bkt], expf(x - rowmax));
    }
    __syncthreads();
    if (tid == 0) {
      const unsigned krank = sh_krank;
      unsigned cum = 0; int sel = 0;
      for (int q = 255; q >= 0; --q) {
        const unsigned c = s_cnt[q];
        if (cum + c >= krank) { sel = q; break; }
        cum += c;
      }
      sh_kpfx  = (kpfx << 8) | (unsigned)sel;
      sh_krank = krank - cum;

      const float prem = sh_prem;
      float fcum = 0.f; int psel = -1;
      for (int q = 255; q >= 0; --q) {
        const float e = s_esum[q];
        if (fcum + e > prem) { psel = q; break; }
        fcum += e;
      }
      if (psel < 0) { psel = 0; fcum -= s_esum[0]; }   // whole subtree kept
      sh_ppfx = (ppfx << 8) | (unsigned)psel;
      sh_prem = prem - fcum;
    }
    __syncthreads();
  }
  const unsigned tauK = sh_kpfx;
  const unsigned tauP = sh_ppfx;
  const unsigned tau  = (tauK > tauP) ? tauK : tauP;

  // ---- partition function over kept set ----
  float zk = 0.f;
  for (int v = tid; v < VV; v += nt) {
    const float x = row[v];
    if (mkey(x) >= tau) zk += expf(x - rowmax);
  }
  redf[tid] = zk;
  __syncthreads();
  for (int s = nt >> 1; s > 0; s >>= 1) { if (tid < s) redf[tid] += redf[tid + s]; __syncthreads(); }
  const float Zk = redf[0];
  const float logZk = logf(Zk);

  // ---- write probs / logprobs ----
  float* __restrict__ prow = probs + (long long)b * VV;
  float* __restrict__ lrow = logprobs + (long long)b * VV;
  for (int v = tid; v < VV; v += nt) {
    const float x = row[v];
    if (mkey(x) >= tau) {
      const float e = expf(x - rowmax);
      prow[v] = e / Zk;
      lrow[v] = (x - rowmax) - logZk;
    } else {
      prow[v] = 0.f;
      lrow[v] = NEGINF;
    }
  }
  if (tid == 0) next_tok[b] = (float)amax;
}

// ---------------------------------------------------------------------------
extern "C" void kernel_launch(void* const* d_in, const int* in_sizes, int n_in,
                              void* d_out, int out_size, void* d_ws, size_t ws_size,
                              hipStream_t stream) {
  const float*     hidden = (const float*)d_in[0];      // [256, 2048] f32
  const float*     embed  = (const float*)d_in[1];      // [50257, 2048] f32
  const long long* lti    = (const long long*)d_in[2];  // [64] i64
  const float*     temps  = (const float*)d_in[3];      // [64] f32
  const float*     tps    = (const float*)d_in[4];      // [64] f32
  const long long* tks    = (const long long*)d_in[5];  // [64] i64

  float* out      = (float*)d_out;
  float* probs    = out;
  float* logprobs = out + (long long)NB * VV;
  float* next_tok = out + 2LL * NB * VV;

  float* h_ws = (float*)d_ws;                     // 64*2048 f32   (512 KB)
  float* lgts = h_ws + (long long)NB * DD;        // 64*50257 f32  (12.9 MB)

  gather_scale_kernel<<<NB, 256, 0, stream>>>(hidden, lti, temps, h_ws);

  const int NWG = (VV + 63) / 64;                 // 786 blocks, 64 vocab cols each
  const size_t smem_bytes = (size_t)4 * TILEF * sizeof(float);  // ~132 KB (of 320 KB/WGP)
  logits_wmma_kernel<<<NWG, 128, smem_bytes, stream>>>(h_ws, embed, lgts);

  sample_kernel<<<NB, 256, 0, stream>>>(lgts, tps, tks, probs, logprobs, next_tok);

  (void)in_sizes; (void)n_in; (void)out_size; (void)ws_size;
}